// RewireGearnet_14800457302375
// MI455X (gfx1250) — compile-verified
//
#include <hip/hip_runtime.h>

#define D  512
#define DOUT 512
#define DE 59
#define NREL 7

typedef float v2f __attribute__((ext_vector_type(2)));
typedef float v8f __attribute__((ext_vector_type(8)));

// ---------------------------------------------------------------------------
// Pack weight matrix W[n, col_off + k] (row stride src_stride, k,n in [0,512))
// into k-pair-interleaved layout: dst[((k>>1)*512 + n)*2 + (k&1)] = W[n][k].
// This makes the WMMA B-operand (4x16 f32: V0 = row k, V1 = row k+1, N = lane)
// loadable as one coalesced float2 per lane.
// ---------------------------------------------------------------------------
__global__ void pack_weights(const float* __restrict__ src, float* __restrict__ dst,
                             int src_stride, int col_off) {
    int idx = blockIdx.x * blockDim.x + threadIdx.x;   // [0, 512*512)
    int bit = idx & 1;
    int p   = idx >> 1;
    int n   = p & (DOUT - 1);
    int k   = ((p >> 9) << 1) | bit;
    dst[idx] = src[(size_t)n * src_stride + col_off + k];
}

__global__ void init_out(const float* __restrict__ b_lin, const float* __restrict__ b_self,
                         float* __restrict__ out) {
    int idx = blockIdx.x * blockDim.x + threadIdx.x;
    int o = idx & (DOUT - 1);
    out[idx] = b_lin[o] + b_self[o];
}

__global__ void zero_buf(float* __restrict__ p, int n4) {
    int idx = blockIdx.x * blockDim.x + threadIdx.x;
    if (idx < n4) ((float4*)p)[idx] = make_float4(0.f, 0.f, 0.f, 0.f);
}

__global__ void relu_inplace(float* __restrict__ out, int n) {
    int idx = blockIdx.x * blockDim.x + threadIdx.x;
    if (idx < n) out[idx] = fmaxf(out[idx], 0.0f);
}

// ---------------------------------------------------------------------------
// Scatter pass for one relation r:
//   agg[node_out[e], :] += w_e * (node_feat[node_in[e], :] + W_edge@ef_e + b_edge)
// for every edge e with relation[e] == r.
// W_edge is cached transposed in LDS ([k][d], 120KB; CDNA5 LDS = 320KB/WGP).
// 256 threads, each thread owns dims d and d+256. agg (41MB) is L2-resident,
// so the float atomics resolve in L2 without touching HBM.
// ---------------------------------------------------------------------------
__global__ void __launch_bounds__(256)
scatter_edges(const float* __restrict__ node_feat, const float* __restrict__ edge_weight,
              const float* __restrict__ edge_feat, const float* __restrict__ W_edge,
              const float* __restrict__ b_edge, const int* __restrict__ node_in,
              const int* __restrict__ node_out, const int* __restrict__ relation,
              float* __restrict__ agg, int rel_target, int E, int edges_per_block) {
    __shared__ float sWT[DE * DOUT];   // [k][d], conflict-free reads (d = lane)
    __shared__ float sEF[64];          // edge feature broadcast
    const int tid = threadIdx.x;

    for (int i = tid; i < DE * DOUT; i += 256) {
        int d = i / DE;
        int k = i - d * DE;
        sWT[k * DOUT + d] = W_edge[i];
    }
    const int d0 = tid, d1 = tid + 256;
    const float bb0 = b_edge[d0];
    const float bb1 = b_edge[d1];
    __syncthreads();

    int e0 = blockIdx.x * edges_per_block;
    int e1 = e0 + edges_per_block;
    if (e1 > E) e1 = E;

    for (int e = e0; e < e1; ++e) {
        if (relation[e] != rel_target) continue;     // block-uniform branch
        __syncthreads();                             // guard sEF WAR vs prev edge
        if (tid < DE) sEF[tid] = edge_feat[(size_t)e * DE + tid];
        __syncthreads();

        const float w  = edge_weight[e];
        const int   ni = node_in[e];
        const int   no = node_out[e];

        float acc0 = bb0, acc1 = bb1;
        #pragma unroll
        for (int k = 0; k < DE; ++k) {
            float ek = sEF[k];
            acc0 = fmaf(sWT[k * DOUT + d0], ek, acc0);
            acc1 = fmaf(sWT[k * DOUT + d1], ek, acc1);
        }
        const float* nf = node_feat + (size_t)ni * D;
        atomicAdd(&agg[(size_t)no * D + d0], w * (nf[d0] + acc0));
        atomicAdd(&agg[(size_t)no * D + d1], w * (nf[d1] + acc1));
    }
}

// ---------------------------------------------------------------------------
// out[M,512] += A[M,512] @ Bpacked  using V_WMMA_F32_16X16X4_F32.
// Wave computes a 32x64 tile: 2 M-subtiles x 4 N-subtiles = 8 accumulators.
// Per K-step: 2 A-loads + 4 B-loads feed 8 WMMAs (0.75 loads/WMMA).
// Block = 8 waves covering 32 rows x 512 cols; grid = M/32 (exact for 20000),
// so EXEC stays all-ones as WMMA requires.
// A-operand (16x4): lanes 0-15 hold rows m, K={k0,k0+1}; lanes 16-31 K={k0+2,k0+3}
//   -> one float2 per lane from row-major A.
// B-operand (4x16): V0 = row ka, V1 = row ka+1, N = lane&15
//   -> one coalesced float2 per lane from the packed layout.
// ---------------------------------------------------------------------------
__global__ void __launch_bounds__(256)
gemm_acc_wmma(const float* __restrict__ A, const float* __restrict__ Bp,
              float* __restrict__ out) {
    const int lane  = threadIdx.x & 31;
    const int wave  = threadIdx.x >> 5;
    const int mlane = lane & 15;
    const int khalf = lane >> 4;                 // 0: K 0/1, 1: K 2/3
    const int row0  = blockIdx.x * 32;
    const int col0  = wave * 64;

    const float* arow0 = A + (size_t)(row0 + mlane) * D;
    const float* arow1 = arow0 + (size_t)16 * D;

    v8f c00 = {}, c01 = {}, c02 = {}, c03 = {};
    v8f c10 = {}, c11 = {}, c12 = {}, c13 = {};

    for (int k0 = 0; k0 < D; k0 += 4) {
        const int ka = k0 + khalf * 2;
        v2f a0 = *(const v2f*)(arow0 + ka);
        v2f a1 = *(const v2f*)(arow1 + ka);
        const float* bb = Bp + (size_t)(ka >> 1) * (DOUT * 2);
        v2f b0 = *(const v2f*)(bb + (size_t)(col0 +  0 + mlane) * 2);
        v2f b1 = *(const v2f*)(bb + (size_t)(col0 + 16 + mlane) * 2);
        v2f b2 = *(const v2f*)(bb + (size_t)(col0 + 32 + mlane) * 2);
        v2f b3 = *(const v2f*)(bb + (size_t)(col0 + 48 + mlane) * 2);
        c00 = __builtin_amdgcn_wmma_f32_16x16x4_f32(false, a0, false, b0, (short)0, c00, false, false);
        c10 = __builtin_amdgcn_wmma_f32_16x16x4_f32(false, a1, false, b0, (short)0, c10, false, false);
        c01 = __builtin_amdgcn_wmma_f32_16x16x4_f32(false, a0, false, b1, (short)0, c01, false, false);
        c11 = __builtin_amdgcn_wmma_f32_16x16x4_f32(false, a1, false, b1, (short)0, c11, false, false);
        c02 = __builtin_amdgcn_wmma_f32_16x16x4_f32(false, a0, false, b2, (short)0, c02, false, false);
        c12 = __builtin_amdgcn_wmma_f32_16x16x4_f32(false, a1, false, b2, (short)0, c12, false, false);
        c03 = __builtin_amdgcn_wmma_f32_16x16x4_f32(false, a0, false, b3, (short)0, c03, false, false);
        c13 = __builtin_amdgcn_wmma_f32_16x16x4_f32(false, a1, false, b3, (short)0, c13, false, false);
    }

    // C/D layout: VGPR v -> M = v + khalf*8, N = lane&15.
    float* ob0 = out + (size_t)(row0 + khalf * 8) * DOUT + col0 + mlane;
    float* ob1 = ob0 + (size_t)16 * DOUT;
    #pragma unroll
    for (int v = 0; v < 8; ++v) {
        float* o0 = ob0 + (size_t)v * DOUT;
        float* o1 = ob1 + (size_t)v * DOUT;
        o0[ 0] += c00[v];
        o0[16] += c01[v];
        o0[32] += c02[v];
        o0[48] += c03[v];
        o1[ 0] += c10[v];
        o1[16] += c11[v];
        o1[32] += c12[v];
        o1[48] += c13[v];
    }
}

// ---------------------------------------------------------------------------
// Orchestration. Workspace: agg (N*512 f32 = 41MB, L2-resident) + packed
// weights ((7+1)*512*512 f32 = 8MB). Total ~49MB.
// ---------------------------------------------------------------------------
extern "C" void kernel_launch(void* const* d_in, const int* in_sizes, int n_in,
                              void* d_out, int out_size, void* d_ws, size_t ws_size,
                              hipStream_t stream) {
    (void)n_in; (void)out_size; (void)ws_size;
    const float* node_feat   = (const float*)d_in[0];
    const float* edge_weight = (const float*)d_in[1];
    const float* edge_feat   = (const float*)d_in[2];
    const float* W_lin       = (const float*)d_in[3];
    const float* b_lin       = (const float*)d_in[4];
    const float* W_self      = (const float*)d_in[5];
    const float* b_self      = (const float*)d_in[6];
    const float* W_edge      = (const float*)d_in[7];
    const float* b_edge      = (const float*)d_in[8];
    const int*   node_in     = (const int*)d_in[9];
    const int*   node_out    = (const int*)d_in[10];
    const int*   relation    = (const int*)d_in[11];

    const int N = in_sizes[0] / D;      // 20000
    const int E = in_sizes[1];          // 640000
    float* out = (float*)d_out;

    float* ws       = (float*)d_ws;
    float* agg      = ws;                                   // N*D
    float* wtp      = ws + (size_t)N * D;                   // NREL*512*512
    float* wtp_self = wtp + (size_t)NREL * DOUT * D;        // 512*512

    const int packBlocks = (DOUT * D) / 256;                // 1024
    for (int r = 0; r < NREL; ++r)
        pack_weights<<<packBlocks, 256, 0, stream>>>(
            W_lin, wtp + (size_t)r * DOUT * D, NREL * D, r * D);
    pack_weights<<<packBlocks, 256, 0, stream>>>(W_self, wtp_self, D, 0);

    init_out<<<(N * DOUT) / 256, 256, 0, stream>>>(b_lin, b_self, out);

    dim3 ggrid(N / 32);                                      // 625
    gemm_acc_wmma<<<ggrid, 256, 0, stream>>>(node_feat, wtp_self, out);

    const int scatterBlocks = 2048;
    const int epb = (E + scatterBlocks - 1) / scatterBlocks;
    const int aggN4 = (N * D) / 4;
    for (int r = 0; r < NREL; ++r) {
        zero_buf<<<(aggN4 + 255) / 256, 256, 0, stream>>>(agg, aggN4);
        scatter_edges<<<scatterBlocks, 256, 0, stream>>>(
            node_feat, edge_weight, edge_feat, W_edge, b_edge,
            node_in, node_out, relation, agg, r, E, epb);
        gemm_acc_wmma<<<ggrid, 256, 0, stream>>>(agg, wtp + (size_t)r * DOUT * D, out);
    }

    relu_inplace<<<(N * DOUT + 255) / 256, 256, 0, stream>>>(out, N * DOUT);
}